// GraphSageEncoder_15522011808351
// MI455X (gfx1250) — compile-verified
//
#include <hip/hip_runtime.h>

typedef __attribute__((ext_vector_type(2))) float v2f;
typedef __attribute__((ext_vector_type(8))) float v8f;

#define DIM 64
#define KPAIRS 32          // 64 K values -> 32 packed pairs
#define LDS_STRIDE 80      // float2 stride per k-pair; 160 dwords = 32 mod 64 banks
                           // -> lower/upper half-wave hit disjoint bank sets

// ---------------------------------------------------------------------------
// Fill agg with -inf (float4 stores)
// ---------------------------------------------------------------------------
__global__ __launch_bounds__(256)
void init_agg_kernel(float4* __restrict__ agg, int n4) {
    int i = blockIdx.x * blockDim.x + threadIdx.x;
    if (i < n4) {
        const float ninf = -__builtin_inff();
        agg[i] = make_float4(ninf, ninf, ninf, ninf);
    }
}

// ---------------------------------------------------------------------------
// Scatter-max: one edge per 32-lane slice; each lane handles 2 features.
// float2 gather from h[src], two native f32 max-atomics into agg[dst].
// agg (25.6MB) stays L2-resident on MI455X (192MB L2).
// ---------------------------------------------------------------------------
__global__ __launch_bounds__(256)
void scatter_max_kernel(const float* __restrict__ h,
                        const long long* __restrict__ ei,
                        float* __restrict__ agg, int nEdges) {
    int t = blockIdx.x * blockDim.x + threadIdx.x;
    int e = t >> 5;
    if (e >= nEdges) return;
    int lane = t & 31;
    int src = (int)ei[e];
    int dst = (int)ei[e + (size_t)nEdges];
    float2 v = ((const float2*)(h + (size_t)src * DIM))[lane];
    float* ap = agg + (size_t)dst * DIM + lane * 2;
    unsafeAtomicMax(ap, v.x);
    unsafeAtomicMax(ap + 1, v.y);
}

// ---------------------------------------------------------------------------
// Fused: out = relu( fixup(agg) @ Wl + bl + h @ Wr )
// One wave computes a 16-row x 64-col strip with V_WMMA_F32_16X16X4_F32.
// Weights staged once per block into LDS in packed K-pair layout:
//   sW[kp*LDS_STRIDE + n] = (W[2kp][n], W[2kp+1][n])  -> one ds_load_b64/frag
// A-frag (16x4 f32): lane<16 -> M=lane, K={k0,k0+1}; lane>=16 -> K={k0+2,k0+3}
// C/D (16x16 f32):  VGPR r -> M = r + 8*(lane>=16), N = lane&15
// ---------------------------------------------------------------------------
__global__ __launch_bounds__(256)
void sage_fused_gemm_kernel(const float* __restrict__ agg,
                            const float* __restrict__ hin,
                            const float* __restrict__ Wl,
                            const float* __restrict__ bl,
                            const float* __restrict__ Wr,
                            float* __restrict__ out, int nRows) {
    __shared__ float2 sWl[KPAIRS * LDS_STRIDE];
    __shared__ float2 sWr[KPAIRS * LDS_STRIDE];
    __shared__ float  sBl[DIM];

    // ---- stage weights + bias (all threads participate; barrier before exit)
    for (int i = threadIdx.x; i < KPAIRS * DIM; i += 256) {
        int kp = i >> 6;        // 0..31
        int n  = i & 63;
        sWl[kp * LDS_STRIDE + n] =
            make_float2(Wl[(2 * kp) * DIM + n], Wl[(2 * kp + 1) * DIM + n]);
        sWr[kp * LDS_STRIDE + n] =
            make_float2(Wr[(2 * kp) * DIM + n], Wr[(2 * kp + 1) * DIM + n]);
    }
    if (threadIdx.x < DIM) sBl[threadIdx.x] = bl[threadIdx.x];
    __syncthreads();

    int wave = blockIdx.x * (blockDim.x >> 5) + (threadIdx.x >> 5);
    int lane = threadIdx.x & 31;
    int m0 = wave * 16;
    if (m0 < nRows) {
        int half = lane >> 4;   // 0: K pair kp, 1: K pair kp+1 within the k-step
        int l16  = lane & 15;

        const float* aggRow = agg + (size_t)(m0 + l16) * DIM;
        const float* hRow   = hin + (size_t)(m0 + l16) * DIM;
        const float ninf = -__builtin_inff();

        v8f acc[4];
        acc[0] = v8f{}; acc[1] = v8f{}; acc[2] = v8f{}; acc[3] = v8f{};

#pragma unroll
        for (int k0 = 0; k0 < DIM; k0 += 4) {
            int ka = k0 + 2 * half;        // even
            int kp = ka >> 1;              // k-pair index this half-wave reads
            float a0 = aggRow[ka];
            float a1 = aggRow[ka + 1];
            v2f aA, aH;
            aA.x = (a0 == ninf) ? 0.0f : a0;   // empty-segment fixup fused here
            aA.y = (a1 == ninf) ? 0.0f : a1;
            aH.x = hRow[ka];
            aH.y = hRow[ka + 1];
#pragma unroll
            for (int nt = 0; nt < 4; ++nt) {
                int n = nt * 16 + l16;
                float2 wl2 = sWl[kp * LDS_STRIDE + n];   // ds_load_b64
                float2 wr2 = sWr[kp * LDS_STRIDE + n];   // ds_load_b64
                v2f bL, bR;
                bL.x = wl2.x; bL.y = wl2.y;
                bR.x = wr2.x; bR.y = wr2.y;
                acc[nt] = __builtin_amdgcn_wmma_f32_16x16x4_f32(
                    false, aA, false, bL, (short)0, acc[nt], false, false);
                acc[nt] = __builtin_amdgcn_wmma_f32_16x16x4_f32(
                    false, aH, false, bR, (short)0, acc[nt], false, false);
            }
        }

        // ---- epilogue: bias + ReLU, store row-major
#pragma unroll
        for (int nt = 0; nt < 4; ++nt) {
            int col = nt * 16 + l16;
            float b = sBl[col];
#pragma unroll
            for (int r = 0; r < 8; ++r) {
                int row = m0 + r + 8 * half;
                float v = acc[nt][r] + b;
                out[(size_t)row * DIM + col] = v > 0.0f ? v : 0.0f;
            }
        }
    }
}

// ---------------------------------------------------------------------------
extern "C" void kernel_launch(void* const* d_in, const int* in_sizes, int n_in,
                              void* d_out, int out_size, void* d_ws, size_t ws_size,
                              hipStream_t stream) {
    const float*     x  = (const float*)d_in[0];
    const long long* ei = (const long long*)d_in[1];   // int64 edge_index [2, E]
    const float*     Wl = (const float*)d_in[2];       // [4, 64, 64]
    const float*     bl = (const float*)d_in[3];       // [4, 64]
    const float*     Wr = (const float*)d_in[4];       // [4, 64, 64]
    float* out = (float*)d_out;

    int nNodes = in_sizes[0] / DIM;
    int nEdges = in_sizes[1] / 2;
    size_t nd = (size_t)nNodes * DIM;

    float* buf0 = (float*)d_ws;
    float* buf1 = buf0 + nd;
    float* agg  = buf1 + nd;

    const float* hin = x;
    for (int layer = 0; layer < 4; ++layer) {
        float* hout = (layer == 3) ? out : ((layer & 1) == 0 ? buf0 : buf1);

        int n4 = (int)(nd / 4);
        init_agg_kernel<<<(n4 + 255) / 256, 256, 0, stream>>>((float4*)agg, n4);

        int threadsS = nEdges * 32;                 // 40M threads, fits int
        scatter_max_kernel<<<(threadsS + 255) / 256, 256, 0, stream>>>(
            hin, ei, agg, nEdges);

        int waves = (nNodes + 15) / 16;
        int blocksG = (waves + 7) / 8;              // 8 waves per 256-thread block
        sage_fused_gemm_kernel<<<blocksG, 256, 0, stream>>>(
            agg, hin,
            Wl + (size_t)layer * DIM * DIM,
            bl + (size_t)layer * DIM,
            Wr + (size_t)layer * DIM * DIM,
            hout, nNodes);

        hin = hout;
    }
}